// VectorQuantization_27504970564158
// MI455X (gfx1250) — compile-verified
//
#include <hip/hip_runtime.h>

typedef float v2f __attribute__((ext_vector_type(2)));
typedef float v8f __attribute__((ext_vector_type(8)));

#define VQ_D      256     // feature dim
#define VQ_K      8192    // codebook size
#define VQ_N      32768   // number of queries (8 * 4096)
#define MTILE     128     // queries per block
#define CTILE     64      // codes per LDS chunk
#define LDA       260     // padded LDS row stride (dwords) -> conflict-free b64 frags
#define NTHREADS  256     // 8 wave32s
#define NCHUNKS   (VQ_K / CTILE)

// ---------------------------------------------------------------------------
// Kernel 1: per-code squared norms  vsq[k] = sum_d v[k][d]^2
// ---------------------------------------------------------------------------
__global__ __launch_bounds__(NTHREADS)
void vq_vsq_kernel(const float* __restrict__ v, float* __restrict__ vsq) {
  int k = blockIdx.x * blockDim.x + threadIdx.x;
  if (k < VQ_K) {
    const float4* row = (const float4*)(v + (size_t)k * VQ_D);
    float s = 0.f;
#pragma unroll 8
    for (int i = 0; i < VQ_D / 4; ++i) {
      float4 f = row[i];
      s = fmaf(f.x, f.x, s);
      s = fmaf(f.y, f.y, s);
      s = fmaf(f.z, f.z, s);
      s = fmaf(f.w, f.w, s);
    }
    vsq[k] = s;
  }
}

// ---------------------------------------------------------------------------
// Kernel 2: fp32 WMMA GEMM (x . v^T) + fused argmin.
//   score[n][k] = vsq[k] - 2 * <x_n, v_k>   ;  out[n] = argmin_k score
// Double-buffered via CDNA5 async DMA: GLOBAL_LOAD_ASYNC_TO_LDS_B128
// (ASYNCcnt) prefetches chunk c+1 while v_wmma_f32_16x16x4_f32 consumes
// chunk c. No staging VGPRs, vsq loads (LOADcnt) never wait on the prefetch.
// ---------------------------------------------------------------------------
__global__ __launch_bounds__(NTHREADS)
void vq_argmin_kernel(const float* __restrict__ x,
                      const float* __restrict__ vecs,
                      const float* __restrict__ vsq,
                      float* __restrict__ out) {
  extern __shared__ float smem[];
  float* xs  = smem;                          // [MTILE][LDA] query tile
  float* vs0 = smem + MTILE * LDA;            // [CTILE][LDA] chunk buffer 0
  float* vs1 = vs0 + CTILE * LDA;             // [CTILE][LDA] chunk buffer 1

  const int tid   = threadIdx.x;
  const int lane  = tid & 31;
  const int wave  = tid >> 5;       // 0..7
  const int col   = lane & 15;      // N-column within a 16x16 tile
  const int hi    = lane >> 4;      // half-select (K interleave / row+8)
  const int qbase = blockIdx.x * MTILE;
  const int qw    = wave * 16;      // this wave's query-row base in the tile

  // Per-thread slice of a chunk: row r0 = tid>>6 (+4 per step), 4 floats @ c4.
  const int r0 = tid >> 6;
  const int c4 = (tid & 63) * 4;
  // Low 32 bits of a generic __shared__ pointer on amdgcn == raw LDS byte addr.
  const unsigned lds_v0 = (unsigned)(size_t)(vs0 + r0 * LDA + c4);
  const unsigned lds_v1 = (unsigned)(size_t)(vs1 + r0 * LDA + c4);

  // Issue this thread's 16 async b128 DMAs for one 64-code chunk.
  auto issue_chunk = [&](int cbase, int buf) {
    unsigned long long g = (unsigned long long)(vecs + (size_t)(cbase + r0) * VQ_D + c4);
    unsigned l = buf ? lds_v1 : lds_v0;
#pragma unroll
    for (int j = 0; j < 16; ++j) {
      asm volatile("global_load_async_to_lds_b128 %0, %1, off"
                   :: "v"(l), "v"(g) : "memory");
      g += 4ull * VQ_D * sizeof(float);   // 4 rows ahead in global (4096 B)
      l += 4u * LDA * sizeof(float);      // 4 rows ahead in LDS    (4160 B)
    }
  };

  // Prefetch chunk 0 into buffer 0 (async engine fills LDS while we stage X).
  issue_chunk(0, 0);

  // --- stage X tile (128 x 256 fp32) into LDS ---
  for (int i = tid; i < MTILE * (VQ_D / 4); i += NTHREADS) {
    int r = i >> 6, c = i & 63;
    float4 f = ((const float4*)(x + (size_t)(qbase + r) * VQ_D))[c];
    *(float4*)&xs[r * LDA + 4 * c] = f;
  }

  asm volatile("s_wait_asynccnt 0x0" ::: "memory");
  __syncthreads();   // X tile + chunk 0 visible to all waves

  float best_s[8];
  int   best_i[8];
#pragma unroll
  for (int r = 0; r < 8; ++r) { best_s[r] = 3.402823466e38f; best_i[r] = 0; }

  // A-fragment base: lane reads x[qw+col][d + 2*hi .. +1]  (ISA 16x4 f32 layout)
  const float* xrow = &xs[(qw + col) * LDA + 2 * hi];
  const int    boff = col * LDA + 2 * hi;

  for (int c = 0; c < NCHUNKS; ++c) {
    const int cbase = c * CTILE;
    const float* vb = (c & 1) ? vs1 : vs0;

    // vsq for THIS chunk (LOADcnt; independent of the async prefetch)
    float q0 = vsq[cbase +  0 + col];
    float q1 = vsq[cbase + 16 + col];
    float q2 = vsq[cbase + 32 + col];
    float q3 = vsq[cbase + 48 + col];

    // async-prefetch next chunk into the other buffer (WAR-safe: all readers
    // of that buffer passed the previous iteration's barrier)
    if (c + 1 < NCHUNKS) issue_chunk(cbase + CTILE, (c + 1) & 1);

    const float* b0p = vb + boff;
    const float* b1p = b0p + 16 * LDA;
    const float* b2p = b0p + 32 * LDA;
    const float* b3p = b0p + 48 * LDA;

    v8f acc0 = {0.f,0.f,0.f,0.f,0.f,0.f,0.f,0.f};
    v8f acc1 = {0.f,0.f,0.f,0.f,0.f,0.f,0.f,0.f};
    v8f acc2 = {0.f,0.f,0.f,0.f,0.f,0.f,0.f,0.f};
    v8f acc3 = {0.f,0.f,0.f,0.f,0.f,0.f,0.f,0.f};

#pragma unroll 4
    for (int d = 0; d < VQ_D; d += 4) {
      v2f a  = *(const v2f*)(xrow + d);
      v2f b0 = *(const v2f*)(b0p + d);
      v2f b1 = *(const v2f*)(b1p + d);
      v2f b2 = *(const v2f*)(b2p + d);
      v2f b3 = *(const v2f*)(b3p + d);
      acc0 = __builtin_amdgcn_wmma_f32_16x16x4_f32(false, a, false, b0, (short)0, acc0, false, false);
      acc1 = __builtin_amdgcn_wmma_f32_16x16x4_f32(false, a, false, b1, (short)0, acc1, false, false);
      acc2 = __builtin_amdgcn_wmma_f32_16x16x4_f32(false, a, false, b2, (short)0, acc2, false, false);
      acc3 = __builtin_amdgcn_wmma_f32_16x16x4_f32(false, a, false, b3, (short)0, acc3, false, false);
    }

    // per-lane running argmin (ascending-index order + strict '<' keeps the
    // lowest index on ties, matching jnp.argmin)
    const int i0 = cbase +  0 + col;
    const int i1 = cbase + 16 + col;
    const int i2 = cbase + 32 + col;
    const int i3 = cbase + 48 + col;
#pragma unroll
    for (int r = 0; r < 8; ++r) {
      float s0 = fmaf(-2.f, acc0[r], q0);
      float s1 = fmaf(-2.f, acc1[r], q1);
      float s2 = fmaf(-2.f, acc2[r], q2);
      float s3 = fmaf(-2.f, acc3[r], q3);
      if (s0 < best_s[r]) { best_s[r] = s0; best_i[r] = i0; }
      if (s1 < best_s[r]) { best_s[r] = s1; best_i[r] = i1; }
      if (s2 < best_s[r]) { best_s[r] = s2; best_i[r] = i2; }
      if (s3 < best_s[r]) { best_s[r] = s3; best_i[r] = i3; }
    }

    // my async DMAs for chunk c+1 are done; barrier makes them visible and
    // confirms every wave is done reading buffer c&1
    asm volatile("s_wait_asynccnt 0x0" ::: "memory");
    __syncthreads();
  }

  // --- one final butterfly min over the 16 columns of each half ---
  // C/D layout: VGPR r -> row r (lanes 0-15) and row r+8 (lanes 16-31)
#pragma unroll
  for (int r = 0; r < 8; ++r) {
    float s = best_s[r];
    int idx = best_i[r];
#pragma unroll
    for (int m = 1; m < 16; m <<= 1) {
      float os = __shfl_xor(s, m, 32);
      int   oi = __shfl_xor(idx, m, 32);
      if (os < s || (os == s && oi < idx)) { s = os; idx = oi; }
    }
    if (col == 0) out[qbase + qw + 8 * hi + r] = (float)idx;
  }
}

// ---------------------------------------------------------------------------
extern "C" void kernel_launch(void* const* d_in, const int* in_sizes, int n_in,
                              void* d_out, int out_size, void* d_ws, size_t ws_size,
                              hipStream_t stream) {
  (void)in_sizes; (void)n_in; (void)out_size; (void)ws_size;
  const float* x    = (const float*)d_in[0];   // [32768, 256]
  const float* vecs = (const float*)d_in[1];   // [8192, 256]
  float* vsq = (float*)d_ws;                   // 8192 floats scratch
  float* out = (float*)d_out;                  // 32768 indices (as float)

  vq_vsq_kernel<<<VQ_K / NTHREADS, NTHREADS, 0, stream>>>(vecs, vsq);

  size_t smem = (size_t)(MTILE + 2 * CTILE) * LDA * sizeof(float);  // ~260 KB of 320 KB
  vq_argmin_kernel<<<VQ_N / MTILE, NTHREADS, smem, stream>>>(x, vecs, vsq, out);
}